// EquivariantLayerNorm_86895778333057
// MI455X (gfx1250) — compile-verified
//
#include <hip/hip_runtime.h>
#include <hip/hip_bf16.h>

#define NB_BLOCKS 10000
#define DH 128
#define DE 64
#define EPS_LN 1e-12f
#define EPS_STD 1e-8f

typedef __attribute__((ext_vector_type(2))) float v2f;
typedef __attribute__((ext_vector_type(8))) float v8f;

// ---------------------------------------------------------------- utilities
__device__ __forceinline__ float wave_sum32(float v) {
    #pragma unroll
    for (int m = 16; m > 0; m >>= 1) v += __shfl_xor(v, m, 32);
    return v;
}

__global__ void zero_kernel(float* __restrict__ p, int n) {
    int i = blockIdx.x * blockDim.x + threadIdx.x;
    if (i < n) p[i] = 0.0f;
}

// block_id is sorted: build block_start[NB+1] (CSR-style segment offsets)
__global__ void block_start_kernel(const int* __restrict__ bid,
                                   int* __restrict__ bstart, int N, int NB) {
    int i = blockIdx.x * blockDim.x + threadIdx.x;
    if (i >= N) return;
    int cur  = bid[i];
    int prev = (i == 0) ? -1 : bid[i - 1];
    for (int b = prev + 1; b <= cur; ++b) bstart[b] = i;
    if (i == N - 1)
        for (int b = cur + 1; b <= NB; ++b) bstart[b] = N;
}

// ------------------------------------------------- H: per-block LN via WMMA
// One workgroup (8 waves) per block. Wave w owns features [16w,16w+16).
// Stats via V_WMMA_F32_16X16X4_F32 with A = ones(16x4):
//   D[m,n] = sum_k B[k,n]  (every output row == column sums; K mapping free)
// Main loop is branch-free (full 4-row chunks); one predicated tail chunk.
__global__ void __launch_bounds__(256) h_block_kernel(
    const float* __restrict__ H, const float* __restrict__ lw,
    const float* __restrict__ lb, const int* __restrict__ bstart,
    float* __restrict__ Hout) {
    __shared__ __align__(16) float smean[DH];
    __shared__ __align__(16) float srstd[DH];
    int b = blockIdx.x;
    int s = bstart[b], e = bstart[b + 1];
    int n = e - s;
    if (n <= 0) return;

    int lane = threadIdx.x & 31;
    int wv   = threadIdx.x >> 5;      // 0..7 -> feature tile
    int fb   = wv * 16;
    int fl   = lane & 15;             // feature within tile (== N column)
    int half = lane >> 4;             // 0/1 -> which pair of K rows

    v8f cs = {};                      // running feature sums
    v8f cq = {};                      // running feature sums of squares
    v2f a;  a.x = 1.0f; a.y = 1.0f;   // ones A-matrix (all lanes)

    const float* __restrict__ col = H + fb + fl;
    int emain = s + (n & ~3);         // full 4-row chunks, no bounds checks

    for (int r0 = s; r0 < emain; r0 += 4) {
        int ra = r0 + half * 2;
        float xa = col[(size_t)ra * DH];
        float xb = col[(size_t)(ra + 1) * DH];
        v2f bv; bv.x = xa;      bv.y = xb;
        v2f bq; bq.x = xa * xa; bq.y = xb * xb;
        cs = __builtin_amdgcn_wmma_f32_16x16x4_f32(false, a, false, bv,
                                                   (short)0, cs, false, false);
        cq = __builtin_amdgcn_wmma_f32_16x16x4_f32(false, a, false, bq,
                                                   (short)0, cq, false, false);
    }
    if (emain < e) {                  // uniform tail: 1..3 rows, zero-padded
        int ra = emain + half * 2;
        int rb = ra + 1;
        float xa = (ra < e) ? col[(size_t)ra * DH] : 0.0f;
        float xb = (rb < e) ? col[(size_t)rb * DH] : 0.0f;
        v2f bv; bv.x = xa;      bv.y = xb;
        v2f bq; bq.x = xa * xa; bq.y = xb * xb;
        cs = __builtin_amdgcn_wmma_f32_16x16x4_f32(false, a, false, bv,
                                                   (short)0, cs, false, false);
        cq = __builtin_amdgcn_wmma_f32_16x16x4_f32(false, a, false, bq,
                                                   (short)0, cq, false, false);
    }
    // Row M=0 of D lives in VGPR0 (c[0]) of lanes 0..15.
    float inv_n = 1.0f / (float)n;
    if (lane < 16) {
        float mu  = cs[0] * inv_n;
        float var = fmaxf(cq[0] * inv_n - mu * mu, 0.0f);
        smean[fb + fl] = mu;
        srstd[fb + fl] = rsqrtf(var + EPS_LN);
    }
    __syncthreads();

    // normalize: b128 path. 32 threads x float4 cover one row; 8 rows / step.
    int fq = (threadIdx.x & 31) * 4;
    int rr = threadIdx.x >> 5;        // 0..7
    float4 w4  = *(const float4*)&lw[fq];
    float4 b4  = *(const float4*)&lb[fq];
    float4 mu4 = *(const float4*)&smean[fq];
    float4 rs4 = *(const float4*)&srstd[fq];
    for (int r = s + rr; r < e; r += 8) {
        const float4 x = *(const float4*)&H[(size_t)r * DH + fq];
        float4 o;
        o.x = w4.x * (x.x - mu4.x) * rs4.x + b4.x;
        o.y = w4.y * (x.y - mu4.y) * rs4.y + b4.y;
        o.z = w4.z * (x.z - mu4.z) * rs4.z + b4.z;
        o.w = w4.w * (x.w - mu4.w) * rs4.w + b4.w;
        *(float4*)&Hout[(size_t)r * DH + fq] = o;
    }
}

// ------------------------------------------------- Z: per-block equivariant
// One wave per block: centroid -> deviation stats over all 3n values ->
// rescale + Z_out, fully fused (block slice stays in cache).
__global__ void __launch_bounds__(32) z_block_kernel(
    const float* __restrict__ Z, const float* __restrict__ sigma,
    const int* __restrict__ bstart, float* __restrict__ Zout,
    float* __restrict__ Rout) {
    int b = blockIdx.x;
    int s = bstart[b], e = bstart[b + 1];
    int n = e - s;
    int lane = threadIdx.x;

    float sx = 0.f, sy = 0.f, sz = 0.f;
    for (int r = s + lane; r < e; r += 32) {
        sx += Z[3 * (size_t)r + 0];
        sy += Z[3 * (size_t)r + 1];
        sz += Z[3 * (size_t)r + 2];
    }
    sx = wave_sum32(sx); sy = wave_sum32(sy); sz = wave_sum32(sz);
    float inv_n = 1.0f / fmaxf((float)n, 1.0f);
    float cx = sx * inv_n, cy = sy * inv_n, cz = sz * inv_n;

    // scatter_std over the 3n deviations (matches reference's vals/ids3)
    float sD = 0.f, sDD = 0.f;
    for (int r = s + lane; r < e; r += 32) {
        float dx = Z[3 * (size_t)r + 0] - cx;
        float dy = Z[3 * (size_t)r + 1] - cy;
        float dz = Z[3 * (size_t)r + 2] - cz;
        sD  += dx + dy + dz;
        sDD += dx * dx + dy * dy + dz * dz;
    }
    sD  = wave_sum32(sD);
    sDD = wave_sum32(sDD);
    float cnt3 = 3.0f * (float)n;
    float m    = sD / fmaxf(cnt3, 1.0f);
    float ss   = sDD - 2.0f * m * sD + cnt3 * m * m;   // == sum((d-m)^2)
    float varE = sqrtf(ss / fmaxf(cnt3 - 1.0f, 1.0f)) + EPS_STD;
    float inv  = 1.0f / varE;

    float r0 = sigma[0] * inv, r1 = sigma[1] * inv, r2 = sigma[2] * inv;
    if (lane < 3) Rout[(size_t)b * 3 + lane] = sigma[lane] * inv;

    for (int r = s + lane; r < e; r += 32) {
        size_t i = 3 * (size_t)r;
        Zout[i + 0] = cx + (Z[i + 0] - cx) * r0;
        Zout[i + 1] = cy + (Z[i + 1] - cy) * r1;
        Zout[i + 2] = cz + (Z[i + 2] - cz) * r2;
    }
}

// ------------------------------------------------- Edges: unsorted segments
// Pass 1: b128 loads, hardware f32 atomics into L2-resident [NB,64] stats.
// 16 edges per workgroup; 16 threads x float4 per edge row.
__global__ void __launch_bounds__(256) edge_stats_kernel(
    const float* __restrict__ EA, const int* __restrict__ eid0,
    const int* __restrict__ bid, float* __restrict__ Esum,
    float* __restrict__ Esq, float* __restrict__ Ecnt, int E) {
    __shared__ int sb[16];
    long long base = (long long)blockIdx.x * 16;
    int t  = threadIdx.x;
    int le = t >> 4;                  // edge slot 0..15
    int fq = (t & 15) * 4;            // feature quad
    long long e = base + le;
    if (e < E) __builtin_prefetch(&EA[(size_t)e * DE + fq], 0, 1);
    if (t < 16) {
        long long ei = base + t;
        int bb = 0;
        if (ei < E) {
            bb = bid[eid0[ei]];
            unsafeAtomicAdd(&Ecnt[bb], 1.0f);
        }
        sb[t] = bb;
    }
    __syncthreads();
    if (e < E) {
        int bb = sb[le];
        float4 v = *(const float4*)&EA[(size_t)e * DE + fq];
        float* ps = &Esum[(size_t)bb * DE + fq];
        float* pq = &Esq [(size_t)bb * DE + fq];
        unsafeAtomicAdd(ps + 0, v.x); unsafeAtomicAdd(pq + 0, v.x * v.x);
        unsafeAtomicAdd(ps + 1, v.y); unsafeAtomicAdd(pq + 1, v.y * v.y);
        unsafeAtomicAdd(ps + 2, v.z); unsafeAtomicAdd(pq + 2, v.z * v.z);
        unsafeAtomicAdd(ps + 3, v.w); unsafeAtomicAdd(pq + 3, v.w * v.w);
    }
}

// Pass 2: normalize each edge row with its segment's stats (all b128).
__global__ void __launch_bounds__(256) edge_norm_kernel(
    const float* __restrict__ EA, const int* __restrict__ eid0,
    const int* __restrict__ bid, const float* __restrict__ Esum,
    const float* __restrict__ Esq, const float* __restrict__ Ecnt,
    const float* __restrict__ lw, const float* __restrict__ lb,
    float* __restrict__ Eout, int E) {
    __shared__ int   sb[16];
    __shared__ float sc[16];
    long long base = (long long)blockIdx.x * 16;
    int t  = threadIdx.x;
    int le = t >> 4;
    int fq = (t & 15) * 4;
    long long e = base + le;
    if (e < E) __builtin_prefetch(&EA[(size_t)e * DE + fq], 0, 1);
    if (t < 16) {
        long long ei = base + t;
        int bb = 0; float c = 1.0f;
        if (ei < E) {
            bb = bid[eid0[ei]];
            c  = fmaxf(Ecnt[bb], 1.0f);
        }
        sb[t] = bb; sc[t] = c;
    }
    __syncthreads();
    if (e < E) {
        int   bb   = sb[le];
        float invc = 1.0f / sc[le];
        float4 s4 = *(const float4*)&Esum[(size_t)bb * DE + fq];
        float4 q4 = *(const float4*)&Esq [(size_t)bb * DE + fq];
        float4 x  = *(const float4*)&EA[(size_t)e * DE + fq];
        float4 w4 = *(const float4*)&lw[fq];
        float4 b4 = *(const float4*)&lb[fq];
        float4 o;
        float mu, var;
        mu = s4.x * invc; var = fmaxf(q4.x * invc - mu * mu, 0.0f);
        o.x = w4.x * (x.x - mu) * rsqrtf(var + EPS_LN) + b4.x;
        mu = s4.y * invc; var = fmaxf(q4.y * invc - mu * mu, 0.0f);
        o.y = w4.y * (x.y - mu) * rsqrtf(var + EPS_LN) + b4.y;
        mu = s4.z * invc; var = fmaxf(q4.z * invc - mu * mu, 0.0f);
        o.z = w4.z * (x.z - mu) * rsqrtf(var + EPS_LN) + b4.z;
        mu = s4.w * invc; var = fmaxf(q4.w * invc - mu * mu, 0.0f);
        o.w = w4.w * (x.w - mu) * rsqrtf(var + EPS_LN) + b4.w;
        *(float4*)&Eout[(size_t)e * DE + fq] = o;
    }
}

// ---------------------------------------------------------------- launcher
extern "C" void kernel_launch(void* const* d_in, const int* in_sizes, int n_in,
                              void* d_out, int out_size, void* d_ws, size_t ws_size,
                              hipStream_t stream) {
    const float* H     = (const float*)d_in[0];
    const float* Z     = (const float*)d_in[1];
    const float* EA    = (const float*)d_in[2];
    const float* sigma = (const float*)d_in[3];
    const float* lnHw  = (const float*)d_in[4];
    const float* lnHb  = (const float*)d_in[5];
    const float* lnEw  = (const float*)d_in[6];
    const float* lnEb  = (const float*)d_in[7];
    const int*   bid   = (const int*)d_in[8];
    const int*   eid0  = (const int*)d_in[9];   // edge_id[0] = first E entries

    const int N  = in_sizes[0] / DH;
    const int E  = in_sizes[2] / DE;
    const int NB = NB_BLOCKS;

    float* out  = (float*)d_out;
    float* Hout = out;
    float* Zout = Hout + (size_t)N * DH;
    float* Eout = Zout + (size_t)N * 3;
    float* Rout = Eout + (size_t)E * DE;

    float* ws     = (float*)d_ws;
    float* Esum   = ws;
    float* Esq    = Esum + (size_t)NB * DE;
    float* Ecnt   = Esq  + (size_t)NB * DE;
    int*   bstart = (int*)(Ecnt + NB);

    int nz = NB * DE * 2 + NB;
    zero_kernel<<<(nz + 255) / 256, 256, 0, stream>>>(ws, nz);
    block_start_kernel<<<(N + 255) / 256, 256, 0, stream>>>(bid, bstart, N, NB);
    h_block_kernel<<<NB, 256, 0, stream>>>(H, lnHw, lnHb, bstart, Hout);
    z_block_kernel<<<NB, 32, 0, stream>>>(Z, sigma, bstart, Zout, Rout);
    edge_stats_kernel<<<(E + 15) / 16, 256, 0, stream>>>(EA, eid0, bid, Esum, Esq, Ecnt, E);
    edge_norm_kernel<<<(E + 15) / 16, 256, 0, stream>>>(EA, eid0, bid, Esum, Esq, Ecnt,
                                                        lnEw, lnEb, Eout, E);
}